// SperimentalSpiralMeshReader_65824668779071
// MI455X (gfx1250) — compile-verified
//
#include <hip/hip_runtime.h>
#include <math.h>

#define NGRAPH 50
#define INDIM 64
#define HID 128
#define MLPH 256
#define ODIM 16
#define GEPS 1e-5f

typedef __attribute__((ext_vector_type(2))) float v2f;
typedef __attribute__((ext_vector_type(8))) float v8f;

// ---------------- utility ----------------
__global__ void fill_kernel(float* p, int n, float v) {
  int i = blockIdx.x * blockDim.x + threadIdx.x;
  int stride = gridDim.x * blockDim.x;
  for (; i < n; i += stride) p[i] = v;
}

__device__ __forceinline__ void atomicMaxF(float* addr, float v) {
  if (v >= 0.f) atomicMax((int*)addr, __float_as_int(v));
  else          atomicMin((unsigned int*)addr, __float_as_uint(v));
}
__device__ __forceinline__ void atomicMinF(float* addr, float v) {
  if (v >= 0.f) atomicMin((int*)addr, __float_as_int(v));
  else          atomicMax((unsigned int*)addr, __float_as_uint(v));
}

// ---------------- 1) degrees ----------------
__global__ void degree_kernel(const int* __restrict__ src, const int* __restrict__ dst,
                              float* outd, float* ind, int E) {
  int e = blockIdx.x * blockDim.x + threadIdx.x;
  if (e < E) {
    atomicAdd(&outd[src[e]], 1.0f);
    atomicAdd(&ind[dst[e]], 1.0f);
  }
}

__global__ void norm_kernel(float* outd, float* ind, int N) {
  int i = blockIdx.x * blockDim.x + threadIdx.x;
  if (i < N) {
    outd[i] = rsqrtf(fmaxf(outd[i], 1.0f));  // becomes out-norm
    ind[i]  = rsqrtf(fmaxf(ind[i], 1.0f));   // becomes in-norm
  }
}

// ---------------- 2) edge gather + scatter-add (64-dim) ----------------
// thread handles 16 consecutive feats of one edge (4 threads / edge)
__global__ void scatter_kernel(const float* __restrict__ feats, const float* __restrict__ ew,
                               const int* __restrict__ src, const int* __restrict__ dst,
                               const float* __restrict__ outn, float* __restrict__ agg, int E) {
  int gid = blockIdx.x * blockDim.x + threadIdx.x;
  if (gid >= E * 4) return;
  int e = gid >> 2;
  int c = (gid & 3) * 16;
  int s = src[e], d = dst[e];
  const float4* fr = (const float4*)(feats + s * INDIM + c);
  __builtin_prefetch(fr, 0, 3);  // global_prefetch_b8: overlap gather with ew/outn loads
  float w = ew[e] * outn[s];
  float* ar = agg + d * INDIM + c;
#pragma unroll
  for (int i = 0; i < 4; ++i) {
    float4 v = fr[i];
    atomicAdd(ar + i * 4 + 0, v.x * w);
    atomicAdd(ar + i * 4 + 1, v.y * w);
    atomicAdd(ar + i * 4 + 2, v.z * w);
    atomicAdd(ar + i * 4 + 3, v.w * w);
  }
}

// ---------------- 3) (agg * in_norm) @ Wconv via V_WMMA_F32_16X16X4_F32 ----------------
// + leaky relu + fused per-graph sum/sumsq/count accumulation (saves a full 51MB pass)
// block = 256 threads = 8 waves; block handles 16 node-rows, wave w handles cols [16w,16w+16)
__global__ __launch_bounds__(256) void conv_wmma_kernel(const float* __restrict__ agg,
                                                        const float* __restrict__ inn,
                                                        const float* __restrict__ W,
                                                        const int* __restrict__ gids,
                                                        float* __restrict__ h,
                                                        float* gsum, float* gsumsq, float* gcnt,
                                                        int N) {
  __shared__ float sW[INDIM * HID];  // 32 KB
  int tid = threadIdx.x;
  {  // vectorized stage: global_load_b128 + ds_store_b128
    const float4* Wg = (const float4*)W;
    float4* Ws = (float4*)sW;
#pragma unroll
    for (int i = 0; i < (INDIM * HID) / (4 * 256); ++i)
      Ws[i * 256 + tid] = Wg[i * 256 + tid];
  }
  __syncthreads();

  int wave = tid >> 5;
  int lane = tid & 31;
  int r = lane & 15;     // A-row / B-col lane index
  int halo = lane >> 4;  // 0: K pair {0,1}, 1: K pair {2,3}
  int colbase = wave * 16;
  int m16 = blockIdx.x * 16;

  int row = m16 + r;
  int crow = row < N ? row : N - 1;  // clamp for tail tiles (reads only)
  const float2* arow2 = (const float2*)(agg + crow * INDIM);
  float nrm = inn[crow];

  v8f c = {0.f, 0.f, 0.f, 0.f, 0.f, 0.f, 0.f, 0.f};
#pragma unroll
  for (int k = 0; k < 16; ++k) {
    int k0 = 4 * k + 2 * halo;           // even -> aligned float2
    float2 av = arow2[k0 >> 1];          // global_load_b64
    v2f a, b;
    a.x = av.x * nrm;
    a.y = av.y * nrm;
    b.x = sW[k0 * HID + colbase + r];
    b.y = sW[(k0 + 1) * HID + colbase + r];
    c = __builtin_amdgcn_wmma_f32_16x16x4_f32(false, a, false, b, (short)0, c, false, false);
  }

  int col = colbase + r;
#pragma unroll
  for (int v = 0; v < 8; ++v) {
    int orow = m16 + v + 8 * halo;  // C layout: VGPR v -> row v (+8 for upper half-wave)
    if (orow < N) {
      float x = c[v];
      x = x > 0.f ? x : 0.01f * x;  // leaky relu
      h[orow * HID + col] = x;
      int g = gids[orow];
      atomicAdd(&gsum[g * HID + col], x);
      atomicAdd(&gsumsq[g * HID + col], x * x);
      // count each node row exactly once: wave 0, col%16==0 lanes (lane 0 & 16)
      if (wave == 0 && r == 0) atomicAdd(&gcnt[g], 1.0f);
    }
  }
}

// ---------------- 4) fold GraphNorm into affine scale/shift per (graph,feature) ----------------
// var = E[(x-a*mu)^2] = E[x^2] - mu^2 * a * (2-a);  hn = scale*x + shift
__global__ void finalize_kernel(const float* gsum, const float* gsumsq, const float* gcnt,
                                const float* alpha, const float* gamma, const float* beta,
                                float* scale, float* shift) {
  int i = blockIdx.x * blockDim.x + threadIdx.x;
  if (i >= NGRAPH * HID) return;
  int g = i / HID, f = i % HID;
  float cnt = fmaxf(gcnt[g], 1.0f);
  float mu = gsum[i] / cnt;
  float ex2 = gsumsq[i] / cnt;
  float a = alpha[f];
  float var = ex2 - mu * mu * a * (2.0f - a);
  float sc = gamma[f] * rsqrtf(var + GEPS);
  scale[i] = sc;
  shift[i] = beta[f] - sc * a * mu;
}

// ---------------- 5) normalize + mean/max/min readout ----------------
__global__ void readout_kernel(const float* __restrict__ h, const int* __restrict__ gids,
                               const float* __restrict__ scale, const float* __restrict__ shift,
                               float* gmean, float* gmax, float* gmin, int N) {
  int gid = blockIdx.x * blockDim.x + threadIdx.x;
  if (gid >= N * (HID / 4)) return;
  int node = gid >> 5;
  int q = (gid & 31) * 4;
  int g = gids[node];
  int base = g * HID + q;
  float4 v = *(const float4*)(h + node * HID + q);
  float4 sc = *(const float4*)(scale + base);
  float4 sf = *(const float4*)(shift + base);
  float x[4] = {sc.x * v.x + sf.x, sc.y * v.y + sf.y, sc.z * v.z + sf.z, sc.w * v.w + sf.w};
#pragma unroll
  for (int j = 0; j < 4; ++j) {
    atomicAdd(gmean + base + j, x[j]);
    atomicMaxF(gmax + base + j, x[j]);
    atomicMinF(gmin + base + j, x[j]);
  }
}

// ---------------- 6) MLP head: one block per graph ----------------
__global__ __launch_bounds__(256) void mlp_kernel(const float* gmean, const float* gmax,
                                                  const float* gmin, const float* gcnt,
                                                  const float* __restrict__ W1, const float* __restrict__ b1,
                                                  const float* __restrict__ W2, const float* __restrict__ b2,
                                                  float* out) {
  __shared__ float sg[3 * HID];
  __shared__ float sh[MLPH];
  int g = blockIdx.x, t = threadIdx.x;
  float cnt = fmaxf(gcnt[g], 1.0f);
  for (int f = t; f < HID; f += 256) {
    sg[f]           = gmean[g * HID + f] / cnt;
    sg[HID + f]     = gmax[g * HID + f];
    sg[2 * HID + f] = gmin[g * HID + f];
  }
  __syncthreads();
  float acc = b1[t];
  for (int k = 0; k < 3 * HID; ++k) acc += sg[k] * W1[k * MLPH + t];
  sh[t] = fmaxf(acc, 0.0f);
  __syncthreads();
  if (t < ODIM) {
    float o = b2[t];
    for (int k = 0; k < MLPH; ++k) o += sh[k] * W2[k * ODIM + t];
    out[g * ODIM + t] = o;
  }
}

extern "C" void kernel_launch(void* const* d_in, const int* in_sizes, int n_in,
                              void* d_out, int out_size, void* d_ws, size_t ws_size,
                              hipStream_t stream) {
  const float* feats = (const float*)d_in[0];
  const float* ew    = (const float*)d_in[1];
  const float* Wconv = (const float*)d_in[2];
  const float* alpha = (const float*)d_in[3];
  const float* gamma = (const float*)d_in[4];
  const float* beta  = (const float*)d_in[5];
  const float* W1    = (const float*)d_in[6];
  const float* b1    = (const float*)d_in[7];
  const float* W2    = (const float*)d_in[8];
  const float* b2    = (const float*)d_in[9];
  const int* src  = (const int*)d_in[10];
  const int* dst  = (const int*)d_in[11];
  const int* gids = (const int*)d_in[12];
  float* out = (float*)d_out;

  const int N = in_sizes[0] / INDIM;   // 100000
  const int E = in_sizes[1];           // 1600000

  // workspace layout (floats)
  float* w = (float*)d_ws;
  float* outn = w;                 // [N]
  float* inn  = w + N;             // [N]
  float* agg  = w + 2 * N;         // [N*64]
  float* h    = agg + (size_t)N * INDIM;  // [N*128]
  float* S    = h + (size_t)N * HID;
  float* gsum   = S;
  float* gsumsq = S + NGRAPH * HID;
  float* gcnt   = S + 2 * NGRAPH * HID;        // 64-slot pad
  float* gmean  = gcnt + 64;
  float* gmax   = gmean + NGRAPH * HID;
  float* gmin   = gmax + NGRAPH * HID;
  float* scale  = gmin + NGRAPH * HID;
  float* shift  = scale + NGRAPH * HID;

  const int TB = 256;
  // init
  fill_kernel<<<(2 * N + TB - 1) / TB, TB, 0, stream>>>(outn, 2 * N, 0.0f);
  fill_kernel<<<(N * INDIM + TB - 1) / TB, TB, 0, stream>>>(agg, N * INDIM, 0.0f);
  fill_kernel<<<(3 * NGRAPH * HID + 64 + TB - 1) / TB, TB, 0, stream>>>(
      gsum, 3 * NGRAPH * HID + 64, 0.0f);  // gsum, gsumsq, gcnt, gmean
  fill_kernel<<<(NGRAPH * HID + TB - 1) / TB, TB, 0, stream>>>(gmax, NGRAPH * HID, -INFINITY);
  fill_kernel<<<(NGRAPH * HID + TB - 1) / TB, TB, 0, stream>>>(gmin, NGRAPH * HID, INFINITY);

  degree_kernel<<<(E + TB - 1) / TB, TB, 0, stream>>>(src, dst, outn, inn, E);
  norm_kernel<<<(N + TB - 1) / TB, TB, 0, stream>>>(outn, inn, N);
  scatter_kernel<<<(E * 4 + TB - 1) / TB, TB, 0, stream>>>(feats, ew, src, dst, outn, agg, E);
  conv_wmma_kernel<<<(N + 15) / 16, 256, 0, stream>>>(agg, inn, Wconv, gids, h,
                                                      gsum, gsumsq, gcnt, N);
  finalize_kernel<<<(NGRAPH * HID + TB - 1) / TB, TB, 0, stream>>>(gsum, gsumsq, gcnt,
                                                                   alpha, gamma, beta, scale, shift);
  readout_kernel<<<(N * (HID / 4) + TB - 1) / TB, TB, 0, stream>>>(h, gids, scale, shift,
                                                                   gmean, gmax, gmin, N);
  mlp_kernel<<<NGRAPH, 256, 0, stream>>>(gmean, gmax, gmin, gcnt, W1, b1, W2, b2, out);
}